// HashGridPositionalEncoding4D_18356690223509
// MI455X (gfx1250) — compile-verified
//
#include <hip/hip_runtime.h>

// HashGridPositionalEncoding4D for MI455X (gfx1250, wave32).
// Memory-bound gather + elementwise add, ~280MB HBM traffic -> ~12us floor.
// x tile staged through LDS with CDNA5 async copies (ASYNCcnt) overlapping
// the hash + table-gather chain; coords/out streamed non-temporal so the
// 8MB hash tables stay resident in the 192MB L2.

typedef float v4f __attribute__((ext_vector_type(4)));
typedef float v2f __attribute__((ext_vector_type(2)));
typedef __attribute__((address_space(3))) v4f lds_v4f;   // 32-bit LDS pointer

#define NUM_LEVELS 16
#define D_MODEL 32          // NUM_LEVELS * LEVEL_DIM
#define TABLE_SIZE 65536    // 16^4 -> mod is a mask
#define THREADS 256

// Exact replication of: h = uint32( fmod(v, 2^32) ), v >= 0, v < ~2^39.
// q = trunc(v * 2^-32) is exact; fma(q, -2^32, v) is exact (result is a
// multiple of ulp(v) with magnitude < 2^32).
__device__ __forceinline__ unsigned int hash_mod32(float v) {
    float q = __builtin_truncf(v * 0x1p-32f);
    float r = __builtin_fmaf(q, -0x1p32f, v);
    return (unsigned int)r;   // v_cvt_u32_f32 truncates, matches astype(uint32)
}

__global__ __launch_bounds__(THREADS)
void hashgrid_enc_kernel(const float* __restrict__ x,
                         const float* __restrict__ coords,
                         const float* __restrict__ tables,
                         const float* __restrict__ freq,
                         float* __restrict__ out,
                         int P)
{
    __shared__ v4f xtile[THREADS * (D_MODEL / 4)];   // 32KB x-tile per block

    // fp32-rounded primes, identical to np.float32(PRIMES)
    const float PR1 = 2654435761.0f;
    const float PR2 = 805459861.0f;
    const float PR3 = 3674653429.0f;

    // freq_bands are uniform -> one s_load_b512, hoisted
    float fb[NUM_LEVELS];
#pragma unroll
    for (int l = 0; l < NUM_LEVELS; ++l) fb[l] = freq[l];

    const int tid    = (int)threadIdx.x;
    const int stride = (int)(gridDim.x * blockDim.x);
    const v2f* __restrict__ tab2 = (const v2f*)tables;

    // This thread's LDS slice as an addrspace(3) pointer: the register value
    // is the 32-bit LDS byte address (what the async engine's VDST wants),
    // and passing it into the asm makes xtile escape so the readback loads
    // below cannot be treated as uninitialized.
    lds_v4f* lbase = (lds_v4f*)&xtile[tid * (D_MODEL / 4)];

    for (int p = (int)(blockIdx.x * blockDim.x) + tid; p < P; p += stride) {
        // ---- Fire-and-forget: stage this point's 128B x slice into LDS via
        //      the gfx1250 async copy path (ASYNCcnt). Runs in the shadow of
        //      the hash + gather chain below. GVS mode: lds_addr, voff32, saddr.
        {
            const unsigned goff = (unsigned)p * (unsigned)(D_MODEL * 4);
#pragma unroll
            for (int i = 0; i < D_MODEL / 4; ++i) {
                unsigned g = goff + (unsigned)(i * 16);
                asm volatile("global_load_async_to_lds_b128 %0, %1, %2"
                             :: "v"(lbase + i), "v"(g), "s"(x)
                             : "memory");
            }
        }

        // Speculative prefetch of next grid-stride iteration's x slice
        // (gfx1250 global_prefetch_b8; dropped silently past end of buffer).
        __builtin_prefetch(x + (size_t)(p + stride) * D_MODEL, 0, 3);

        // ---- coords (streamed once -> non-temporal b128) ----
        v4f c = __builtin_nontemporal_load((const v4f*)coords + p);
        float c0 = __builtin_fminf(__builtin_fmaxf(c.x, 0.0f), 1.0f);
        float c1 = __builtin_fminf(__builtin_fmaxf(c.y, 0.0f), 1.0f);
        float c2 = __builtin_fminf(__builtin_fmaxf(c.z, 0.0f), 1.0f);
        float c3 = __builtin_fminf(__builtin_fmaxf(c.w, 0.0f), 1.0f);

        // ---- 16 hash computations + table gathers (L2-resident, default TH) ----
        float e[D_MODEL];
#pragma unroll
        for (int l = 0; l < NUM_LEVELS; ++l) {
            float f = fb[l];
            // prime[0] == 1.0f and c0*f <= 128 < 2^32, so fmod is the identity
            unsigned int h0 = (unsigned int)(c0 * f);
            unsigned int h1 = hash_mod32((c1 * f) * PR1);
            unsigned int h2 = hash_mod32((c2 * f) * PR2);
            unsigned int h3 = hash_mod32((c3 * f) * PR3);
            unsigned int idx = (h0 ^ h1 ^ h2 ^ h3) & (TABLE_SIZE - 1u);
            v2f t = tab2[(size_t)((unsigned)l << 16) + idx];  // global_load_b64
            e[2 * l + 0] = t.x;
            e[2 * l + 1] = t.y;
        }

        // ---- drain the async engine, then LDS readback + add + NT store ----
        asm volatile("s_wait_asynccnt 0x0" ::: "memory");

        const v4f* __restrict__ xl = &xtile[tid * (D_MODEL / 4)];
        v4f* __restrict__ op = (v4f*)out + (size_t)p * (D_MODEL / 4);
#pragma unroll
        for (int i = 0; i < D_MODEL / 4; ++i) {
            v4f xv = xl[i];                     // ds_load_b128
            v4f r;
            r.x = xv.x + e[4 * i + 0];
            r.y = xv.y + e[4 * i + 1];
            r.z = xv.z + e[4 * i + 2];
            r.w = xv.w + e[4 * i + 3];
            __builtin_nontemporal_store(r, op + i);
        }
    }
}

extern "C" void kernel_launch(void* const* d_in, const int* in_sizes, int n_in,
                              void* d_out, int out_size, void* d_ws, size_t ws_size,
                              hipStream_t stream) {
    (void)n_in; (void)d_ws; (void)ws_size; (void)out_size;
    const float* x      = (const float*)d_in[0];  // (B, N, 32)
    const float* coords = (const float*)d_in[1];  // (B, N, 4)
    const float* tables = (const float*)d_in[2];  // (16, 65536, 2)
    const float* freq   = (const float*)d_in[3];  // (16,)
    float* out = (float*)d_out;                   // (B, N, 32)

    const int P = in_sizes[0] / D_MODEL;          // B*N points

    const int threads = THREADS;                  // 8 wave32 waves per block
    const int blocks  = 2048;                     // grid-stride: 2 points/thread at P=1M
    hipLaunchKernelGGL(hashgrid_enc_kernel, dim3(blocks), dim3(threads), 0, stream,
                       x, coords, tables, freq, out, P);
}